// SpatialEncoder_70600672412128
// MI455X (gfx1250) — compile-verified
//
#include <hip/hip_runtime.h>
#include <hip/hip_bf16.h>

#define DD 64
#define LN_EPS 1e-5f
#define WP_STRIDE 160   // floats per K-pair row in LDS; +160 ≡ +32 banks -> conflict-free

typedef float v2f __attribute__((ext_vector_type(2)));
typedef float v8f __attribute__((ext_vector_type(8)));

// ---------------------------------------------------------------------------
// Degree / normalization kernels
// ---------------------------------------------------------------------------
__global__ void deg_init_kernel(float* deg, int n) {
    int i = blockIdx.x * blockDim.x + threadIdx.x;
    if (i < n) deg[i] = 1.0f;  // self-loop contributes 1 to every node's degree
}

__global__ void deg_edges_kernel(const int* __restrict__ col, float* deg, int e) {
    int i = blockIdx.x * blockDim.x + threadIdx.x;
    if (i < e) atomicAdd(&deg[col[i]], 1.0f);
}

__global__ void dinv_kernel(float* deg_dinv, int n) {
    int i = blockIdx.x * blockDim.x + threadIdx.x;
    if (i < n) deg_dinv[i] = rsqrtf(deg_dinv[i]);  // deg >= 1 always (self-loop)
}

// ---------------------------------------------------------------------------
// WMMA GEMM: HW[r][n] = sum_k H[r][k] * W[k][n]   (H: nrows x 64, W: 64 x 64)
// One wave -> 16 rows x 64 cols via V_WMMA_F32_16X16X4_F32, K stepped by 4.
// A fragment (16x4 f32): lane<16 holds M=lane, K=k0+0/1; lane>=16 -> K=k0+2/3.
// B fragments are staged in LDS pre-paired by K so each fragment is a single
// aligned ds_load_b64:  sWp[(k/2)*WP_STRIDE + n*2 + (k&1)] = W[k][n].
// C/D (16x16 f32): vgpr v, lanes 0-15 -> (M=v, N=lane); 16-31 -> (M=v+8).
// ---------------------------------------------------------------------------
__global__ void __launch_bounds__(256)
gemm_wmma_kernel(const float* __restrict__ H, const float* __restrict__ W,
                 float* __restrict__ HW, int nrows) {
    __shared__ float sWp[32 * WP_STRIDE];  // 20 KB of the 320 KB WGP LDS
    int tid = threadIdx.x;
    for (int idx = tid; idx < DD * DD; idx += 256) {
        int k = idx >> 6;        // 0..63
        int n = idx & 63;
        sWp[(k >> 1) * WP_STRIDE + n * 2 + (k & 1)] = W[idx];  // coalesced global read
    }
    __syncthreads();

    int wave = tid >> 5;
    int lane = tid & 31;
    int rowBlock = blockIdx.x * 8 + wave;
    int r0 = rowBlock * 16;
    if (r0 >= nrows) return;  // wave-uniform: EXEC stays all-ones for WMMA

    int lhalf = lane >> 4;   // 0 or 1
    int lmod  = lane & 15;

    v8f acc0 = {}, acc1 = {}, acc2 = {}, acc3 = {};
    const float* hrow = H + (size_t)(r0 + lmod) * DD + 2 * lhalf;

    for (int k0 = 0; k0 < DD; k0 += 4) {
        v2f a = *(const v2f*)(hrow + k0);  // 8B-aligned (k0 even, lhalf*2 even)

        const float* bbase = sWp + ((k0 >> 1) + lhalf) * WP_STRIDE + lmod * 2;
        v2f b0 = *(const v2f*)(bbase +  0);   // N tile 0..15
        v2f b1 = *(const v2f*)(bbase + 32);   // N tile 16..31
        v2f b2 = *(const v2f*)(bbase + 64);   // N tile 32..47
        v2f b3 = *(const v2f*)(bbase + 96);   // N tile 48..63

        acc0 = __builtin_amdgcn_wmma_f32_16x16x4_f32(false, a, false, b0, (short)0, acc0, false, false);
        acc1 = __builtin_amdgcn_wmma_f32_16x16x4_f32(false, a, false, b1, (short)0, acc1, false, false);
        acc2 = __builtin_amdgcn_wmma_f32_16x16x4_f32(false, a, false, b2, (short)0, acc2, false, false);
        acc3 = __builtin_amdgcn_wmma_f32_16x16x4_f32(false, a, false, b3, (short)0, acc3, false, false);
    }

    float* out = HW + (size_t)r0 * DD;
#pragma unroll
    for (int v = 0; v < 8; ++v) {
        int rr = v + 8 * lhalf;
        out[(size_t)rr * DD + lmod +  0] = acc0[v];
        out[(size_t)rr * DD + lmod + 16] = acc1[v];
        out[(size_t)rr * DD + lmod + 32] = acc2[v];
        out[(size_t)rr * DD + lmod + 48] = acc3[v];
    }
}

// ---------------------------------------------------------------------------
// Aggregation init: agg[i][:] = bias + hw[i][:] * dinv[i]^2   (self-loop term)
// One thread per float2.
// ---------------------------------------------------------------------------
__global__ void agg_init_kernel(const float* __restrict__ hw,
                                const float* __restrict__ dinv,
                                const float* __restrict__ bias,
                                float* __restrict__ agg, int n) {
    int idx = blockIdx.x * blockDim.x + threadIdx.x;  // over n*32
    if (idx >= n * 32) return;
    int node = idx >> 5;
    int d2   = (idx & 31) * 2;
    float di = dinv[node];
    float s  = di * di;
    float2 v = *(const float2*)(hw + (size_t)node * DD + d2);
    float2 b = *(const float2*)(bias + d2);
    float2 o;
    o.x = b.x + v.x * s;
    o.y = b.y + v.y * s;
    *(float2*)(agg + (size_t)node * DD + d2) = o;
}

// ---------------------------------------------------------------------------
// Edge scatter: one wave per edge; lane holds float2 of the 64-wide message.
// agg[col] += hw[row] * dinv[row]*dinv[col]
// ---------------------------------------------------------------------------
__global__ void __launch_bounds__(256)
scatter_kernel(const float* __restrict__ hw, const float* __restrict__ dinv,
               const int* __restrict__ row, const int* __restrict__ col,
               float* agg, int e) {
    int widx = (blockIdx.x * blockDim.x + threadIdx.x) >> 5;
    int lane = threadIdx.x & 31;
    if (widx >= e) return;
    int r = row[widx];
    int c = col[widx];
    float en = dinv[r] * dinv[c];
    float2 v = *(const float2*)(hw + (size_t)r * DD + lane * 2);
    float* dst = agg + (size_t)c * DD + lane * 2;
    atomicAdd(dst + 0, v.x * en);
    atomicAdd(dst + 1, v.y * en);
}

// ---------------------------------------------------------------------------
// LayerNorm (+ optional ReLU): one wave per node, wave32 shuffle reduction.
// ---------------------------------------------------------------------------
__global__ void __launch_bounds__(256)
layernorm_kernel(const float* __restrict__ h, const float* __restrict__ gamma,
                 const float* __restrict__ beta, float* __restrict__ out,
                 int n, int do_relu) {
    int node = (blockIdx.x * blockDim.x + threadIdx.x) >> 5;
    int lane = threadIdx.x & 31;
    if (node >= n) return;
    float2 v = *(const float2*)(h + (size_t)node * DD + lane * 2);
    float s = v.x + v.y;
#pragma unroll
    for (int off = 16; off > 0; off >>= 1) s += __shfl_xor(s, off, 32);
    float mu = s * (1.0f / 64.0f);
    float dx = v.x - mu, dy = v.y - mu;
    float q = dx * dx + dy * dy;
#pragma unroll
    for (int off = 16; off > 0; off >>= 1) q += __shfl_xor(q, off, 32);
    float rstd = rsqrtf(q * (1.0f / 64.0f) + LN_EPS);
    float2 g = *(const float2*)(gamma + lane * 2);
    float2 b = *(const float2*)(beta + lane * 2);
    float2 o;
    o.x = dx * rstd * g.x + b.x;
    o.y = dy * rstd * g.y + b.y;
    if (do_relu) { o.x = fmaxf(o.x, 0.0f); o.y = fmaxf(o.y, 0.0f); }
    *(float2*)(out + (size_t)node * DD + lane * 2) = o;
}

// ---------------------------------------------------------------------------
// Host launcher
// ---------------------------------------------------------------------------
extern "C" void kernel_launch(void* const* d_in, const int* in_sizes, int n_in,
                              void* d_out, int out_size, void* d_ws, size_t ws_size,
                              hipStream_t stream) {
    const float* x      = (const float*)d_in[0];
    const float* Ws     = (const float*)d_in[1];
    const float* bs     = (const float*)d_in[2];
    const float* gammas = (const float*)d_in[3];
    const float* betas  = (const float*)d_in[4];
    const int*   ei     = (const int*)d_in[5];

    const int N = in_sizes[0] / DD;
    const int L = in_sizes[1] / (DD * DD);
    const int E = in_sizes[5] / 2;
    const int* row = ei;       // sources (gather)
    const int* col = ei + E;   // targets (scatter)

    // Workspace layout (floats): dinv[N] | hw[N*64] | hcur[N*64]
    float* dinv = (float*)d_ws;
    float* hw   = dinv + ((N + 63) & ~63);
    float* hcur = hw + (size_t)N * DD;
    float* hout = (float*)d_out;

    const int TB = 256;

    // Degrees -> dinv (with self-loops)
    deg_init_kernel<<<(N + TB - 1) / TB, TB, 0, stream>>>(dinv, N);
    deg_edges_kernel<<<(E + TB - 1) / TB, TB, 0, stream>>>(col, dinv, E);
    dinv_kernel<<<(N + TB - 1) / TB, TB, 0, stream>>>(dinv, N);

    const int rowBlocks  = (N + 15) / 16;
    const int gemmBlocks = (rowBlocks + 7) / 8;
    const int vecBlocks  = ((N * 32) + TB - 1) / TB;       // agg_init / layernorm waves
    const int edgeWaveBlocks = ((E * 32) + TB - 1) / TB;   // 51.2M threads < 2^31

    for (int l = 0; l < L; ++l) {
        const float* hin = (l == 0) ? x : hcur;
        const float* W = Ws + (size_t)l * DD * DD;
        const float* b = bs + (size_t)l * DD;
        const float* g = gammas + (size_t)l * DD;
        const float* be = betas + (size_t)l * DD;

        gemm_wmma_kernel<<<gemmBlocks, TB, 0, stream>>>(hin, W, hw, N);
        agg_init_kernel<<<vecBlocks, TB, 0, stream>>>(hw, dinv, b, hcur, N);
        scatter_kernel<<<edgeWaveBlocks, TB, 0, stream>>>(hw, dinv, row, col, hcur, E);
        float* lnOut = (l == L - 1) ? hout : hcur;  // in-place LN is safe per-wave
        layernorm_kernel<<<vecBlocks, TB, 0, stream>>>(hcur, g, be, lnOut, N, l < L - 1 ? 1 : 0);
    }
}